// S4D_69097433858260
// MI455X (gfx1250) — compile-verified
//
#include <hip/hip_runtime.h>
#include <math.h>

#define B_  16
#define H_  256
#define L_  4096
#define N2_ 32

#define SEGS   8          // segments per sequence (parallel scan phase split)
#define SEGLEN 512        // L / SEGS
#define CHUNKS 16         // 32-step chunks per segment
#define KW     96         // K width of chunk GEMM: 32 (U) + 64 (G re/im)
#define GSZ    (64*16)    // G state tile: 64 rows (re/im x 32 modes) x 16 batches

typedef __attribute__((ext_vector_type(2))) float v2f;
typedef __attribute__((ext_vector_type(8))) float v8f;

__device__ __forceinline__ v8f wmma4(v2f a, v2f b, v8f c) {
    return __builtin_amdgcn_wmma_f32_16x16x4_f32(false, a, false, b, (short)0, c,
                                                 false, false);
}

// ---------------------------------------------------------------------------
// Kernel 1: per-h chunk transfer tables (one wave32 per h, lane = mode n).
//   A_h (96x96 row-major):
//     rows  0..31 (y):  cols 0..31  T[t][m]   = k[t-m] (lower-tri Toeplitz)
//                       cols 32..63 Re(Ct w^{t+1}); cols 64..95 -Im(Ct w^{t+1})
//     rows 32..63 (Gr): cols 0..31  Re(w_n^{31-m}); col 32+n Re(w32); col 64+n -Im(w32)
//     rows 64..95 (Gi): cols 0..31  Im(w_n^{31-m}); col 32+n Im(w32); col 64+n  Re(w32)
//   w512tab: w^512 per (h,n) for the cross-segment carry combine.
// ---------------------------------------------------------------------------
__global__ void __launch_bounds__(256)
s4d_build_tables(const float* __restrict__ log_dt,
                 const float* __restrict__ C_real,
                 const float* __restrict__ C_imag,
                 const float* __restrict__ log_A_real,
                 const float* __restrict__ A_imag,
                 float* __restrict__ Atab,
                 float* __restrict__ w512tab) {
    const int lane = threadIdx.x & 31;
    const int h = blockIdx.x * 8 + (threadIdx.x >> 5);
    const int idx = (h << 5) | lane;

    float dt = expf(log_dt[h]);
    float Ar = -expf(log_A_real[idx]);
    float Ai = A_imag[idx];
    float e  = expf(dt * Ar);
    float wr = e * cosf(dt * Ai);
    float wi = e * sinf(dt * Ai);
    // Ct = 2 * C * (w - 1) / A
    float nr = wr - 1.0f, ni = wi;
    float inv = 1.0f / (Ar * Ar + Ai * Ai);
    float qr = (nr * Ar + ni * Ai) * inv;
    float qi = (ni * Ar - nr * Ai) * inv;
    float cr = C_real[idx], ci = C_imag[idx];
    float ctr = 2.0f * (cr * qr - ci * qi);
    float cti = 2.0f * (cr * qi + ci * qr);

    float* A = Atab + (size_t)h * (KW * KW);
    // zero the table (lane owns 3 rows), then order vs. the scattered fills
    const float4 z4 = make_float4(0.f, 0.f, 0.f, 0.f);
    for (int r = 0; r < 3; ++r) {
        float* rp = A + (3 * lane + r) * KW;
        for (int q = 0; q < KW; q += 4) *(float4*)&rp[q] = z4;
    }
    asm volatile("s_wait_storecnt 0x0" ::: "memory");

    float pr = 1.0f, pi = 0.0f;                 // w^p running power
    for (int p = 0; p <= 32; ++p) {
        if (p < 32) {
            // P block: w^{31-m} at column m = 31-p
            A[(32 + lane) * KW + (31 - p)] = pr;
            A[(64 + lane) * KW + (31 - p)] = pi;
            // k[p] = sum_n Re(Ct_n w_n^p): butterfly over modes
            float kc = ctr * pr - cti * pi;
            kc += __shfl_xor(kc, 16, 32);
            kc += __shfl_xor(kc,  8, 32);
            kc += __shfl_xor(kc,  4, 32);
            kc += __shfl_xor(kc,  2, 32);
            kc += __shfl_xor(kc,  1, 32);
            // Toeplitz diagonal t-m == p : lane writes (t = p+lane, m = lane)
            if (p + lane < 32) A[(p + lane) * KW + lane] = kc;
        }
        if (p >= 1) {                            // carry rows, t = p-1
            float mr = ctr * pr - cti * pi;      // Re(Ct w^p)
            float mi = ctr * pi + cti * pr;      // Im(Ct w^p)
            A[(p - 1) * KW + 32 + lane] = mr;
            A[(p - 1) * KW + 64 + lane] = -mi;
        }
        if (p == 32) {                           // Q block: w^32 rotation
            A[(32 + lane) * KW + 32 + lane] = pr;
            A[(32 + lane) * KW + 64 + lane] = -pi;
            A[(64 + lane) * KW + 32 + lane] = pi;
            A[(64 + lane) * KW + 64 + lane] = pr;
            float sr = pr, si = pi;              // w^512 = (w^32)^16
            for (int q = 0; q < 4; ++q) {
                float tr = sr * sr - si * si;
                float ti = 2.0f * sr * si;
                sr = tr; si = ti;
            }
            w512tab[h * 64 + lane]      = sr;
            w512tab[h * 64 + 32 + lane] = si;
        }
        float npr = pr * wr - pi * wi;
        float npi = pr * wi + pi * wr;
        pr = npr; pi = npi;
    }
}

// ---------------------------------------------------------------------------
// Kernels 2 & 4: WMMA chunk scan. Block = 8 waves = the 8 segments of one h;
// the 36KB A_h table is staged once into LDS and shared. Each wave carries
// G (64x16: re/im modes x batches) in its private LDS slab, and per 32-step
// chunk runs one K=96 WMMA loop producing 6 tiles:
//   [y0,y1]   = [T|M] @ [U;G]        (EMIT_Y only)
//   [G0..G3]  = [P|Q] @ [U;G]        (state advance)
// Phase A (EMIT_Y=0): G starts at zero, final G -> Gout (segment-local state).
// Phase C (EMIT_Y=1): G starts from combined carries; emits D*u + exact GELU.
// ---------------------------------------------------------------------------
template <bool EMIT_Y>
__global__ void __launch_bounds__(256)
s4d_chunk_scan(const float* __restrict__ u,
               const float* __restrict__ Atab,
               const float* __restrict__ carryG,
               const float* __restrict__ D,
               float* __restrict__ Gout,
               float* __restrict__ ybuf) {
    extern __shared__ float smem[];
    float* Alds = smem;                          // 96*96 floats
    const int tid  = threadIdx.x;
    const int warp = tid >> 5;
    const int lane = tid & 31;
    const int h = blockIdx.x;
    const int s = warp;                          // segment id
    float* G = smem + KW * KW + warp * GSZ;

    const float* Ah = Atab + (size_t)h * (KW * KW);
    for (int i = tid; i < (KW * KW) / 4; i += 256)          // 9 uniform iters
        *(float4*)&Alds[i * 4] = *(const float4*)&Ah[i * 4];

    if (EMIT_Y) {
        const float* cg = carryG + (size_t)(h * SEGS + s) * GSZ;
        for (int q = 0; q < 8; ++q) {
            int ix = (q * 32 + lane) * 4;
            *(float4*)&G[ix] = *(const float4*)&cg[ix];
        }
    } else {
        const float4 z4 = make_float4(0.f, 0.f, 0.f, 0.f);
        for (int q = 0; q < 8; ++q) {
            int ix = (q * 32 + lane) * 4;
            *(float4*)&G[ix] = z4;
        }
    }
    __syncthreads();

    const int cl = lane & 15;                    // frag row (A) == col (B/C)
    const int k0 = (lane >> 4) << 1;             // frag K offset
    const int rb = (lane >> 4) << 3;             // C-tile row base
    const float Dh = EMIT_Y ? D[h] : 0.0f;
    const float* ubase = u + ((size_t)cl * H_ + h) * (size_t)L_;  // batch cl

#pragma unroll 1
    for (int c = 0; c < CHUNKS; ++c) {
        const int l0 = s * SEGLEN + c * 32;
        __builtin_prefetch(ubase + l0 + 32, 0, 1);           // next chunk of u
        v8f aY0 = {0,0,0,0,0,0,0,0}, aY1 = {0,0,0,0,0,0,0,0};
        v8f aG0 = {0,0,0,0,0,0,0,0}, aG1 = {0,0,0,0,0,0,0,0};
        v8f aG2 = {0,0,0,0,0,0,0,0}, aG3 = {0,0,0,0,0,0,0,0};
#pragma unroll
        for (int k = 0; k < KW; k += 4) {
            const int kk = k + k0;
            v2f bf;
            if (k < 32) {                        // S rows 0..31: u chunk
                bf.x = ubase[l0 + kk];
                bf.y = ubase[l0 + kk + 1];
            } else {                             // S rows 32..95: state G
                bf.x = G[(kk - 32) * 16 + cl];
                bf.y = G[(kk - 31) * 16 + cl];
            }
            if (EMIT_Y) {
                aY0 = wmma4(*(const v2f*)&Alds[(     cl) * KW + kk], bf, aY0);
                aY1 = wmma4(*(const v2f*)&Alds[(16 + cl) * KW + kk], bf, aY1);
            }
            aG0 = wmma4(*(const v2f*)&Alds[(32 + cl) * KW + kk], bf, aG0);
            aG1 = wmma4(*(const v2f*)&Alds[(48 + cl) * KW + kk], bf, aG1);
            aG2 = wmma4(*(const v2f*)&Alds[(64 + cl) * KW + kk], bf, aG2);
            aG3 = wmma4(*(const v2f*)&Alds[(80 + cl) * KW + kk], bf, aG3);
        }
        // advance state: C-layout accumulators -> row-major G in LDS
#pragma unroll
        for (int j = 0; j < 8; ++j) {
            G[( 0 + rb + j) * 16 + cl] = aG0[j];
            G[(16 + rb + j) * 16 + cl] = aG1[j];
            G[(32 + rb + j) * 16 + cl] = aG2[j];
            G[(48 + rb + j) * 16 + cl] = aG3[j];
        }
        if (EMIT_Y) {                            // skip + exact GELU + store
            const float* ub = ubase + l0;
            float* yb = ybuf + ((size_t)cl * H_ + h) * (size_t)L_ + l0;
#pragma unroll
            for (int yt = 0; yt < 2; ++yt) {
                const v8f acc = yt ? aY1 : aY0;
                const int tb = yt * 16 + rb;     // timestep base, 8 consecutive
                float4 u0 = *(const float4*)&ub[tb];
                float4 u1 = *(const float4*)&ub[tb + 4];
                float4 o0, o1;
#pragma unroll
                for (int j = 0; j < 8; ++j) {
                    float uu = (j < 4) ? (&u0.x)[j] : (&u1.x)[j - 4];
                    float yv = fmaf(Dh, uu, acc[j]);
                    float g  = 0.5f * yv * (1.0f + erff(yv * 0.70710678118654752f));
                    if (j < 4) (&o0.x)[j] = g; else (&o1.x)[j - 4] = g;
                }
                *(float4*)&yb[tb]     = o0;
                *(float4*)&yb[tb + 4] = o1;
            }
        }
    }
    if (!EMIT_Y) {                               // dump segment-local state
        float* go = Gout + (size_t)(h * SEGS + s) * GSZ;
        for (int q = 0; q < 8; ++q) {
            int ix = (q * 32 + lane) * 4;
            *(float4*)&go[ix] = *(const float4*)&G[ix];
        }
    }
}

// ---------------------------------------------------------------------------
// Kernel 3: cross-segment carry combine. One thread per (h, mode, batch):
//   X[s] = w512 * X[s-1] + localG[s];  carry[s] = X[s-1]   (8 serial steps)
// ---------------------------------------------------------------------------
__global__ void __launch_bounds__(256)
s4d_combine(const float* __restrict__ w512tab,
            const float* __restrict__ localG,
            float* __restrict__ carryG) {
    const int tid = blockIdx.x * 256 + threadIdx.x;   // H*N2*B = 131072
    const int b = tid & 15;
    const int n = (tid >> 4) & 31;
    const int h = tid >> 9;
    const float ar = w512tab[h * 64 + n];
    const float ai = w512tab[h * 64 + 32 + n];
    float Xr = 0.0f, Xi = 0.0f;
    for (int s = 0; s < SEGS; ++s) {
        const size_t base = (size_t)(h * SEGS + s) * GSZ;
        carryG[base + n * 16 + b]        = Xr;
        carryG[base + (32 + n) * 16 + b] = Xi;
        float lr = localG[base + n * 16 + b];
        float li = localG[base + (32 + n) * 16 + b];
        float nXr = ar * Xr - ai * Xi + lr;
        float nXi = ai * Xr + ar * Xi + li;
        Xr = nXr; Xi = nXi;
    }
}

// ---------------------------------------------------------------------------
// Kernel 5: 1x1 conv (2H x H GEMM) + bias + GLU, fp32 WMMA 16x16x4.
// One wave per GLU tile-pair (rows m0 / m0+256) sharing the Y B-fragment;
// K=256 -> 64 WMMA per accumulator; GLU fused into the store.
// ---------------------------------------------------------------------------
__global__ void __launch_bounds__(256)
s4d_out_gemm_kernel(const float* __restrict__ Wt,   // (2H, H) row-major
                    const float* __restrict__ bo,   // (2H)
                    const float* __restrict__ y,    // (B, H, L)
                    float* __restrict__ out) {      // (B, H, L)
    const int lane = threadIdx.x & 31;
    const int job  = blockIdx.x * 8 + (threadIdx.x >> 5);
    const int ntile = job & (L_ / 16 - 1);
    const int mpair = (job >> 8) & (H_ / 16 - 1);
    const int b     = job >> 12;
    const int m0 = mpair << 4;
    const int n0 = ntile << 4;
    const int row = lane & 15;
    const int k0  = (lane >> 4) << 1;

    const float* __restrict__ Yb = y + (size_t)b * ((size_t)H_ * L_);
    const float* __restrict__ wa = Wt + (size_t)(m0 + row) * H_;
    const float* __restrict__ wb = Wt + (size_t)(H_ + m0 + row) * H_;

    v8f acca = {0,0,0,0,0,0,0,0};
    v8f accb = {0,0,0,0,0,0,0,0};
#pragma unroll 4
    for (int k = 0; k < H_; k += 4) {
        const int kk = k + k0;
        v2f bf = { Yb[(size_t)kk * L_ + n0 + row],
                   Yb[(size_t)(kk + 1) * L_ + n0 + row] };
        acca = wmma4(*(const v2f*)(wa + kk), bf, acca);
        accb = wmma4(*(const v2f*)(wb + kk), bf, accb);
    }
    const int rbase = (lane >> 4) << 3;
#pragma unroll
    for (int j = 0; j < 8; ++j) {
        const int r = rbase + j;
        float za = acca[j] + bo[m0 + r];
        float zb = accb[j] + bo[H_ + m0 + r];
        float o  = za / (1.0f + __expf(-zb));     // za * sigmoid(zb)
        out[((size_t)b * H_ + (m0 + r)) * (size_t)L_ + n0 + (lane & 15)] = o;
    }
}

// ---------------------------------------------------------------------------
extern "C" void kernel_launch(void* const* d_in, const int* in_sizes, int n_in,
                              void* d_out, int out_size, void* d_ws, size_t ws_size,
                              hipStream_t stream) {
    (void)in_sizes; (void)n_in; (void)out_size; (void)ws_size;
    const float* u       = (const float*)d_in[0];
    const float* log_dt  = (const float*)d_in[1];
    const float* C_real  = (const float*)d_in[2];
    const float* C_imag  = (const float*)d_in[3];
    const float* log_A   = (const float*)d_in[4];
    const float* A_imag  = (const float*)d_in[5];
    const float* D       = (const float*)d_in[6];
    const float* W_out   = (const float*)d_in[7];
    const float* b_out   = (const float*)d_in[8];
    float* out = (float*)d_out;

    char* ws = (char*)d_ws;
    float* Atab    = (float*)(ws + 0);                   //  9.0 MB (256 x 96x96)
    float* w512tab = (float*)(ws + ((size_t)10 << 20));  //  64 KB
    float* localG  = (float*)(ws + ((size_t)11 << 20));  //  8 MB
    float* carryG  = (float*)(ws + ((size_t)20 << 20));  //  8 MB
    float* ybuf    = (float*)(ws + ((size_t)32 << 20));  // 64 MB post-GELU y

    const size_t lds = (size_t)(KW * KW + 8 * GSZ) * sizeof(float);  // 68 KB

    s4d_build_tables<<<H_ / 8, 256, 0, stream>>>(
        log_dt, C_real, C_imag, log_A, A_imag, Atab, w512tab);

    s4d_chunk_scan<false><<<H_, 256, lds, stream>>>(
        u, Atab, carryG, D, localG, ybuf);               // phase A: local states

    s4d_combine<<<(H_ * N2_ * B_) / 256, 256, 0, stream>>>(
        w512tab, localG, carryG);                        // phase B: carries

    s4d_chunk_scan<true><<<H_, 256, lds, stream>>>(
        u, Atab, carryG, D, localG, ybuf);               // phase C: emit y

    s4d_out_gemm_kernel<<<(B_ * (H_ / 16) * (L_ / 16)) / 8, 256, 0, stream>>>(
        W_out, b_out, ybuf, out);
}